// GrNet_15444702397277
// MI455X (gfx1250) — compile-verified
//
#include <hip/hip_runtime.h>

typedef float v2f __attribute__((ext_vector_type(2)));
typedef float v8f __attribute__((ext_vector_type(8)));
typedef unsigned int v4u __attribute__((ext_vector_type(4)));
typedef int v4i __attribute__((ext_vector_type(4)));
typedef int v8i __attribute__((ext_vector_type(8)));

// D = A(16x4,f32) * B(4x16,f32) + C(16x16,f32)   -> v_wmma_f32_16x16x4_f32
__device__ __forceinline__ v8f wmma_f32_16x16x4(v2f a, v2f b, v8f c) {
  return __builtin_amdgcn_wmma_f32_16x16x4_f32(false, a, false, b, (short)0, c,
                                               false, false);
}

// TDM: async-copy `nelem` fp32 elements (one 1-row tile) global -> LDS.
// D# layout per CDNA5 ISA (group0: count=1,type=2;
// group1: data_size=4B, tensor_dim0=tile_dim0=nelem, stride=nelem).
__device__ __forceinline__ void tdm_load_to_lds(const void* gptr, void* lptr,
                                                unsigned int nelem) {
#if defined(__HIP_DEVICE_COMPILE__)
  unsigned long long ga = (unsigned long long)gptr;
  unsigned int lds_off = (unsigned int)(unsigned long long)lptr; // flat LDS: low 32 bits = offset
  v4u g0;
  g0.x = 1u;                                        // count=1 (valid), no gather
  g0.y = lds_off;                                   // lds_addr
  g0.z = (unsigned int)(ga & 0xffffffffu);          // global_addr[31:0]
  g0.w = (unsigned int)((ga >> 32) & 0x01ffffffu)   // global_addr[56:32]
         | (2u << 30);                              // type=2 ("image")
  v8i g1;
  g1[0] = (int)(2u << 16);                          // data_size=4B; wg_mask=0
  g1[1] = (int)((nelem & 0xffffu) << 16);           // tensor_dim0[15:0]
  g1[2] = (int)(((nelem >> 16) & 0xffffu)           // tensor_dim0[31:16]
         | (1u << 16));                             // tensor_dim1=1 (low bits)
  g1[3] = (int)((nelem & 0xffffu) << 16);           // tile_dim0 = nelem
  g1[4] = 1;                                        // tile_dim1=1
  g1[5] = (int)nelem;                               // tensor_dim0_stride[31:0]
  g1[6] = 0;
  g1[7] = 0;
  v4i z4 = {};
#if __clang_major__ >= 23
  v8i z8 = {};
  __builtin_amdgcn_tensor_load_to_lds(g0, g1, z4, z4, z8, 0);
#else
  __builtin_amdgcn_tensor_load_to_lds(g0, g1, z4, z4, 0);
#endif
  __builtin_amdgcn_s_wait_tensorcnt(0);
#else
  (void)gptr; (void)lptr; (void)nelem;
#endif
}

// ---------------- Kernel 1: row-pool W (256x512 -> 128x512) -----------------
__global__ void pool_w_kernel(const float* __restrict__ W, float* __restrict__ Wp) {
  int idx = blockIdx.x * blockDim.x + threadIdx.x;   // 0..65535
  int r = idx >> 9;                                   // 0..127
  int k = idx & 511;
  Wp[idx] = 0.5f * (W[(2 * r) * 512 + k] + W[(2 * r + 1) * 512 + k]);
}

// ---------------- Kernel 2: Z[b] = Wp(128x512) @ X[b](512x16) ---------------
// One block per batch, 8 waves; wave m computes the 16-row tile [16m,16m+16).
// X[b] (32 KB) staged into LDS by the Tensor Data Mover (wave 0 issues).
__global__ void frmap_kernel(const float* __restrict__ Wp,
                             const float* __restrict__ X,
                             float* __restrict__ Z) {
  __shared__ float Xs[512 * 16];     // 32 KB
  const int b = blockIdx.x;
  const int t = threadIdx.x;
  const int wave = t >> 5;
  const int lane = t & 31;

#if defined(__HIP_DEVICE_COMPILE__)
  if (wave == 0) {
    tdm_load_to_lds(X + (size_t)b * (512 * 16), Xs, 512 * 16);  // waits TENSORcnt
  }
#else
  const float4* xg = (const float4*)(X + (size_t)b * (512 * 16));
  float4* xs = (float4*)Xs;
  for (int i = 0; i < 8; ++i) xs[t + 256 * i] = xg[t + 256 * i];
#endif
  __syncthreads();

  const int n = lane & 15;
  const int h = lane >> 4;           // K-half select
  const int row = wave * 16 + n;     // A row for this lane
  const float* wr = Wp + row * 512;

  v8f c = {};
#pragma unroll 4
  for (int k0 = 0; k0 < 512; k0 += 4) {
    const int k = k0 + 2 * h;
    v2f a = *(const v2f*)(wr + k);                    // A[m][k], A[m][k+1]
    v2f bb;
    bb.x = Xs[k * 16 + n];                            // B[k][n]
    bb.y = Xs[(k + 1) * 16 + n];                      // B[k+1][n]
    c = wmma_f32_16x16x4(a, bb, c);
  }
  float* zb = Z + (size_t)b * 2048;
#pragma unroll
  for (int r = 0; r < 8; ++r)
    zb[(wave * 16 + r + 8 * h) * 16 + n] = c[r];
}

// -------- Kernel 3: G=ZtZ, chol, U = Z L^-T (orthonormal), P2 = U Ut --------
__global__ void orth_proj_kernel(const float* __restrict__ Z, float* __restrict__ P2) {
  __shared__ float Zs[128 * 16];
  __shared__ float Us[128 * 16];
  __shared__ float G[256];
  __shared__ float Lm[256];
  __shared__ float Li[256];

  const int b = blockIdx.x;
  const int t = threadIdx.x;
  const int wave = t >> 5, lane = t & 31;
  const int n = lane & 15, h = lane >> 4;

  const float4* zg = (const float4*)(Z + (size_t)b * 2048);
  float4* zs = (float4*)Zs;
#pragma unroll
  for (int i = 0; i < 2; ++i) zs[t + 256 * i] = zg[t + 256 * i];
  __syncthreads();

  // --- G = Z^T Z (16x16, K=128). A-frag == B-frag by symmetry. wave 0 only.
  if (wave == 0) {
    v8f c = {};
#pragma unroll 4
    for (int k0 = 0; k0 < 128; k0 += 4) {
      const int k = k0 + 2 * h;
      v2f a;
      a.x = Zs[k * 16 + n];
      a.y = Zs[(k + 1) * 16 + n];
      c = wmma_f32_16x16x4(a, a, c);
    }
#pragma unroll
    for (int r = 0; r < 8; ++r) G[(r + 8 * h) * 16 + n] = c[r];
  }
  __syncthreads();

  // --- Cholesky G = L L^T and Li = L^-1 (single thread; ~2 KFLOP)
  if (t == 0) {
    for (int j = 0; j < 16; ++j) {
      float s = G[j * 16 + j];
      for (int k = 0; k < j; ++k) s -= Lm[j * 16 + k] * Lm[j * 16 + k];
      float d = sqrtf(fmaxf(s, 1e-24f));
      Lm[j * 16 + j] = d;
      float inv = 1.0f / d;
      for (int i = j + 1; i < 16; ++i) {
        float s2 = G[i * 16 + j];
        for (int k = 0; k < j; ++k) s2 -= Lm[i * 16 + k] * Lm[j * 16 + k];
        Lm[i * 16 + j] = s2 * inv;
      }
    }
    for (int cI = 0; cI < 16; ++cI) {
      for (int i = 0; i < 16; ++i) Li[i * 16 + cI] = 0.0f;
      Li[cI * 16 + cI] = 1.0f / Lm[cI * 16 + cI];
      for (int i = cI + 1; i < 16; ++i) {
        float s = 0.0f;
        for (int k = cI; k < i; ++k) s += Lm[i * 16 + k] * Li[k * 16 + cI];
        Li[i * 16 + cI] = -s / Lm[i * 16 + i];
      }
    }
  }
  __syncthreads();

  // --- U = Z @ L^-T  (128x16 @ 16x16).  B[k][n] = Li[n][k].
  {
    const int row = wave * 16 + n;
    v8f c = {};
#pragma unroll
    for (int k0 = 0; k0 < 16; k0 += 4) {
      const int k = k0 + 2 * h;
      v2f a = *(const v2f*)&Zs[row * 16 + k];
      v2f bb = *(const v2f*)&Li[n * 16 + k];
      c = wmma_f32_16x16x4(a, bb, c);
    }
#pragma unroll
    for (int r = 0; r < 8; ++r) Us[(wave * 16 + r + 8 * h) * 16 + n] = c[r];
  }
  __syncthreads();

  // --- P2 = U U^T (128x128, K=16). wave m owns tile-row m; loop 8 tile-cols.
  float* pb = P2 + (size_t)b * 16384;
  for (int tn = 0; tn < 8; ++tn) {
    v8f c = {};
#pragma unroll
    for (int k0 = 0; k0 < 16; k0 += 4) {
      const int k = k0 + 2 * h;
      v2f a = *(const v2f*)&Us[(wave * 16 + n) * 16 + k];   // A[m][k..k+1]
      v2f bb = *(const v2f*)&Us[(tn * 16 + n) * 16 + k];    // B[k][j]=U[j][k]
      c = wmma_f32_16x16x4(a, bb, c);
    }
#pragma unroll
    for (int r = 0; r < 8; ++r)
      pb[(wave * 16 + r + 8 * h) * 128 + tn * 16 + n] = c[r];
  }
}

// ------- Kernel 4: logits = P2flat(256x16384) @ fc_w^T(16384x100) + b -------
// One block per (mt,nt) tile: 8 waves split K=16384 into 2048-chunks, then a
// deterministic LDS reduction (each thread sums the 8 partial 16x16 tiles in order).
__global__ void fc_kernel(const float* __restrict__ P2,
                          const float* __restrict__ fcw,
                          const float* __restrict__ fcb,
                          float* __restrict__ out) {
  __shared__ float red[8 * 256];
  const int t = threadIdx.x;
  const int wave = t >> 5;
  const int lane = t & 31;
  const int n = lane & 15, h = lane >> 4;
  const int mt = blockIdx.x / 7;   // 0..15 (batch tile)
  const int nt = blockIdx.x % 7;   // 0..6  (class tile)
  const int row = mt * 16 + n;
  const int col = nt * 16 + n;
  const int colc = (col < 100) ? col : 0;  // clamp: garbage only pollutes unstored cols

  const float* arow = P2 + (size_t)row * 16384;
  const float* brow = fcw + (size_t)colc * 16384;
  const int kbase = wave * 2048;

  __builtin_prefetch(arow + kbase, 0, 1);   // global_prefetch_b8
  __builtin_prefetch(brow + kbase, 0, 1);

  v8f c = {};
#pragma unroll 4
  for (int k0 = kbase; k0 < kbase + 2048; k0 += 4) {
    const int k = k0 + 2 * h;
    v2f a = *(const v2f*)(arow + k);
    v2f bb = *(const v2f*)(brow + k);
    c = wmma_f32_16x16x4(a, bb, c);
  }
#pragma unroll
  for (int r = 0; r < 8; ++r)
    red[wave * 256 + (r + 8 * h) * 16 + n] = c[r];
  __syncthreads();

  // thread t owns output element (M = t/16, N = t%16) of this 16x16 tile
  float v = 0.0f;
#pragma unroll
  for (int w = 0; w < 8; ++w) v += red[w * 256 + t];
  const int orow = mt * 16 + (t >> 4);
  const int ocol = nt * 16 + (t & 15);
  if (ocol < 100) out[orow * 100 + ocol] = v + fcb[ocol];
}

extern "C" void kernel_launch(void* const* d_in, const int* in_sizes, int n_in,
                              void* d_out, int out_size, void* d_ws, size_t ws_size,
                              hipStream_t stream) {
  const float* X   = (const float*)d_in[0];   // (256, 512, 16)
  const float* W   = (const float*)d_in[1];   // (256, 512)
  const float* fcw = (const float*)d_in[2];   // (100, 16384)
  const float* fcb = (const float*)d_in[3];   // (100,)
  float* out = (float*)d_out;                 // (256, 100)

  float* Wp = (float*)d_ws;                   // 128*512           (256 KB)
  float* Z  = Wp + 128 * 512;                 // 256*128*16        (2 MB)
  float* P2 = Z + 256 * 128 * 16;             // 256*128*128       (16 MB)

  pool_w_kernel<<<256, 256, 0, stream>>>(W, Wp);
  frmap_kernel<<<256, 256, 0, stream>>>(Wp, X, Z);
  orth_proj_kernel<<<256, 256, 0, stream>>>(Z, P2);
  fc_kernel<<<112, 256, 0, stream>>>(P2, fcw, fcb, out);
}